// Net_7937099563014
// MI455X (gfx1250) — compile-verified
//
#include <hip/hip_runtime.h>
#include <hip/hip_bf16.h>
#include <math.h>

typedef __attribute__((ext_vector_type(2))) float v2f;
typedef __attribute__((ext_vector_type(8))) float v8f;

#define FEAT_F 500
#define FEAT_H 16
#define FEAT_C 40
#define FEAT_CP 48   // W2 column dim padded to 3 WMMA tiles

// ---------------- init ----------------
__global__ void k_init_deg(float* __restrict__ deg, int n) {
    int i = blockIdx.x * blockDim.x + threadIdx.x;
    if (i < n) deg[i] = 1.0f;   // self-loop weight 1 folded into degree
}
__global__ void k_zero(float* __restrict__ p, long n) {
    long i = (long)blockIdx.x * blockDim.x + threadIdx.x;
    if (i < n) p[i] = 0.0f;
}

// ---------------- degree / rsqrt ----------------
__global__ void k_deg_accum(const long long* __restrict__ ei,
                            const float* __restrict__ w,
                            float* __restrict__ deg, int E) {
    int e = blockIdx.x * blockDim.x + threadIdx.x;
    if (e < E) {
        int c = (int)ei[(long)E + e];          // col = targets
        atomicAdd(&deg[c], w[e]);
    }
}
__global__ void k_dinv(float* __restrict__ deg, int n) {
    int i = blockIdx.x * blockDim.x + threadIdx.x;
    if (i < n) {
        float d = deg[i];
        deg[i] = (d > 0.0f) ? rsqrtf(d) : 0.0f;   // store dinv in place
    }
}

// ---------------- GEMM1: h1[N,16] = x[N,500] @ W1[500,16] (WMMA f32 16x16x4) ----
__global__ __launch_bounds__(256) void k_gemm1(const float* __restrict__ x,
                                               const float* __restrict__ W1,
                                               float* __restrict__ h1,
                                               int n_rows) {
    __shared__ float lw[FEAT_F * FEAT_H];                 // 32 KB of the 320 KB LDS
    for (int i = threadIdx.x; i < FEAT_F * FEAT_H; i += blockDim.x) lw[i] = W1[i];
    __syncthreads();

    const int wave = threadIdx.x >> 5;
    const int lane = threadIdx.x & 31;
    const int tile = blockIdx.x * 8 + wave;               // one 16-row tile per wave
    const int ntiles = n_rows >> 4;                       // N is a multiple of 16
    if (tile >= ntiles) return;                           // wave-uniform: EXEC stays all-1s

    const int row0 = tile << 4;
    const int hl   = lane & 15;                           // A row / B col within tile
    const int koff = (lane >> 4) << 1;                    // lanes 0-15: K+0,1; 16-31: K+2,3
    const float* xrow = x + (long)(row0 + hl) * FEAT_F;

    v8f acc = {};
    for (int k0 = 0; k0 < FEAT_F; k0 += 4) {
        v2f a = *(const v2f*)(xrow + k0 + koff);          // 8B-aligned (stride 2000B)
        v2f b;
        b.x = lw[(k0 + koff)     * FEAT_H + hl];
        b.y = lw[(k0 + koff + 1) * FEAT_H + hl];
        acc = __builtin_amdgcn_wmma_f32_16x16x4_f32(false, a, false, b,
                                                    (short)0, acc, false, false);
    }
    // C/D layout: VGPR v -> row v (lanes 0-15) or row v+8 (lanes 16-31), N = lane&15
    const int rbase = row0 + ((lane >> 4) << 3);
#pragma unroll
    for (int v = 0; v < 8; ++v)
        h1[(long)(rbase + v) * FEAT_H + hl] = acc[v];
}

// ---------------- edge aggregation (one thread per edge*feature) ----------------
template <int F>
__global__ void k_agg(const long long* __restrict__ ei,
                      const float* __restrict__ w,
                      const float* __restrict__ dinv,
                      const float* __restrict__ hin,
                      float* __restrict__ aggout, int E) {
    long idx = (long)blockIdx.x * blockDim.x + threadIdx.x;
    long tot = (long)E * F;
    if (idx >= tot) return;
    int e = (int)(idx / F);
    int f = (int)(idx % F);
    int r = (int)ei[e];
    int c = (int)ei[(long)E + e];
    float coef = dinv[r] * w[e] * dinv[c];
    atomicAdd(&aggout[(long)c * F + f], coef * hin[(long)r * F + f]);
}

// ---------------- h = relu(agg1 + dinv^2*h1 + b1), in place ----------------
__global__ void k_relu_self(float* __restrict__ agg1,       // becomes h
                            const float* __restrict__ h1,
                            const float* __restrict__ dinv,
                            const float* __restrict__ b1, long n16) {
    long idx = (long)blockIdx.x * blockDim.x + threadIdx.x;
    if (idx >= n16) return;
    int i = (int)(idx >> 4);
    int f = (int)(idx & 15);
    float di = dinv[i];
    float v = agg1[idx] + di * di * h1[idx] + b1[f];
    agg1[idx] = fmaxf(v, 0.0f);
}

// ---------------- GEMM2: h2[N,40] = h[N,16] @ W2[16,40] (WMMA f32 16x16x4) ------
__global__ __launch_bounds__(256) void k_gemm2(const float* __restrict__ h,
                                               const float* __restrict__ W2,
                                               float* __restrict__ h2,
                                               int n_rows) {
    __shared__ float lw[FEAT_H * FEAT_CP];                // padded 16x48
    for (int i = threadIdx.x; i < FEAT_H * FEAT_CP; i += blockDim.x) {
        int r = i / FEAT_CP, c = i % FEAT_CP;
        lw[i] = (c < FEAT_C) ? W2[r * FEAT_C + c] : 0.0f;
    }
    __syncthreads();

    const int wave = threadIdx.x >> 5;
    const int lane = threadIdx.x & 31;
    const int gtile = blockIdx.x * 8 + wave;
    const int nrt = n_rows >> 4;
    if (gtile >= nrt * 3) return;                         // wave-uniform exit
    const int rt = gtile / 3;
    const int ct = gtile % 3;
    const int row0 = rt << 4;
    const int cbase = ct << 4;
    const int hl = lane & 15;
    const int koff = (lane >> 4) << 1;
    const float* hrow = h + (long)(row0 + hl) * FEAT_H;

    v8f acc = {};
#pragma unroll
    for (int k0 = 0; k0 < FEAT_H; k0 += 4) {
        v2f a = *(const v2f*)(hrow + k0 + koff);
        v2f b;
        b.x = lw[(k0 + koff)     * FEAT_CP + cbase + hl];
        b.y = lw[(k0 + koff + 1) * FEAT_CP + cbase + hl];
        acc = __builtin_amdgcn_wmma_f32_16x16x4_f32(false, a, false, b,
                                                    (short)0, acc, false, false);
    }
    const int col = cbase + hl;
    if (col < FEAT_C) {
        const int rbase = row0 + ((lane >> 4) << 3);
#pragma unroll
        for (int v = 0; v < 8; ++v)
            h2[(long)(rbase + v) * FEAT_C + col] = acc[v];
    }
}

// ---------------- final: z = agg2 + dinv^2*h2 + b2; out = log_softmax(z) --------
__global__ void k_final(const float* __restrict__ agg2,
                        const float* __restrict__ h2,
                        const float* __restrict__ dinv,
                        const float* __restrict__ b2,
                        float* __restrict__ out, int n) {
    int i = blockIdx.x * blockDim.x + threadIdx.x;
    if (i >= n) return;
    float di = dinv[i];
    float di2 = di * di;
    float v[FEAT_C];
    float mx = -INFINITY;
#pragma unroll
    for (int f = 0; f < FEAT_C; ++f) {
        float t = agg2[(long)i * FEAT_C + f] + di2 * h2[(long)i * FEAT_C + f] + b2[f];
        v[f] = t;
        mx = fmaxf(mx, t);
    }
    float s = 0.0f;
#pragma unroll
    for (int f = 0; f < FEAT_C; ++f) s += __expf(v[f] - mx);
    float ls = __logf(s);
#pragma unroll
    for (int f = 0; f < FEAT_C; ++f)
        out[(long)i * FEAT_C + f] = v[f] - mx - ls;
}

extern "C" void kernel_launch(void* const* d_in, const int* in_sizes, int n_in,
                              void* d_out, int out_size, void* d_ws, size_t ws_size,
                              hipStream_t stream) {
    const float*      x  = (const float*)d_in[0];
    const long long*  ei = (const long long*)d_in[1];   // int64 [2,E]
    const float*      w  = (const float*)d_in[2];
    const float*      W1 = (const float*)d_in[3];
    const float*      b1 = (const float*)d_in[4];
    const float*      W2 = (const float*)d_in[5];
    const float*      b2 = (const float*)d_in[6];

    const int E = in_sizes[2];                  // 3,200,000
    const int H = in_sizes[4];                  // 16
    const int C = in_sizes[6];                  // 40
    const int F = in_sizes[3] / H;              // 500
    const int N = in_sizes[0] / F;              // 100,000

    float* ws   = (float*)d_ws;
    float* dinv = ws;                           // N      (deg -> dinv in place)
    float* h1   = dinv + N;                     // 16N
    float* agg1 = h1 + (long)N * H;             // 16N    (becomes h after relu)
    float* h2   = agg1 + (long)N * H;           // 40N
    float* agg2 = h2 + (long)N * C;             // 40N

    const int T = 256;
    auto blk = [](long n, int t) { return (unsigned)((n + t - 1) / t); };

    // init
    k_init_deg<<<blk(N, T), T, 0, stream>>>(dinv, N);
    k_zero<<<blk((long)N * H, T), T, 0, stream>>>(agg1, (long)N * H);
    k_zero<<<blk((long)N * C, T), T, 0, stream>>>(agg2, (long)N * C);

    // degree + dinv
    k_deg_accum<<<blk(E, T), T, 0, stream>>>(ei, w, dinv, E);
    k_dinv<<<blk(N, T), T, 0, stream>>>(dinv, N);

    // layer 1
    const int ntiles1 = N >> 4;                         // N % 16 == 0
    k_gemm1<<<blk(ntiles1, 8), T, 0, stream>>>(x, W1, h1, N);
    k_agg<FEAT_H><<<blk((long)E * H, T), T, 0, stream>>>(ei, w, dinv, h1, agg1, E);
    k_relu_self<<<blk((long)N * H, T), T, 0, stream>>>(agg1, h1, dinv, b1, (long)N * H);

    // layer 2
    k_gemm2<<<blk((long)ntiles1 * 3, 8), T, 0, stream>>>(agg1 /*=h*/, W2, h2, N);
    k_agg<FEAT_C><<<blk((long)E * C, T), T, 0, stream>>>(ei, w, dinv, h2, agg2, E);

    // self-loop + bias + log-softmax -> d_out
    k_final<<<blk(N, T), T, 0, stream>>>(agg2, h2, dinv, b2, (float*)d_out, N);
}